// PrimalDualStraight_15058155340038
// MI455X (gfx1250) — compile-verified
//
#include <hip/hip_runtime.h>
#include <stddef.h>

typedef float v2f __attribute__((ext_vector_type(2)));
typedef float v4f __attribute__((ext_vector_type(4)));
typedef float v8f __attribute__((ext_vector_type(8)));

#define HW  96
#define PIX 9216   // 96*96

// ---------------------------------------------------------------------------
// Forward projection: ev1_part[chunk, a, x] = sum_{p in chunk} S_norm[a,p,x]*img[p]
// img[p] = psq[ch=1][h,w] with p = w*96 + h (fortran flatten of transposed image)
// Block: (a, chunk) ; 288 threads = 24 x-groups (x4 floats) x 12 p-lanes.
// Per step the block reads 2x 4608 contiguous bytes of S_norm (NT B128 stream),
// two accumulators to keep two loads in flight and break the FMA chain.
// ---------------------------------------------------------------------------
__global__ void gemv_forward(const float* __restrict__ S_norm,
                             const float* __restrict__ psq,
                             float* __restrict__ ev1_part) {
    __shared__ float img[PIX];
    __shared__ float red[12 * HW];
    const int t = threadIdx.x;          // 0..287
    const int a = blockIdx.x;           // 0..95
    const int chunk = blockIdx.y;       // 0..3
    for (int idx = t; idx < PIX; idx += 288) {
        int h = idx % HW, w = idx / HW;                 // idx = w*96+h
        img[idx] = psq[PIX + h * HW + w];               // channel 1 of primal
    }
    __syncthreads();
    const int xg = t % 24;
    const int ps = t / 24;              // 0..11
    const int x0 = xg * 4;
    v4f acc0 = {0.f, 0.f, 0.f, 0.f};
    v4f acc1 = {0.f, 0.f, 0.f, 0.f};
    const int pbase = chunk * 2304;
    const float* Sa = S_norm + (size_t)a * PIX * HW;
    for (int pstep = 0; pstep < 2304; pstep += 24) {
        int p0 = pbase + pstep + ps;
        int p1 = p0 + 12;
        v4f s0 = __builtin_nontemporal_load((const v4f*)(Sa + (size_t)p0 * HW + x0));
        v4f s1 = __builtin_nontemporal_load((const v4f*)(Sa + (size_t)p1 * HW + x0));
        acc0 += s0 * img[p0];
        acc1 += s1 * img[p1];
    }
    acc0 += acc1;
    *((v4f*)&red[ps * HW + x0]) = acc0;
    __syncthreads();
    if (t < HW) {
        float sum = 0.f;
        #pragma unroll
        for (int i = 0; i < 12; ++i) sum += red[i * HW + t];
        ev1_part[(size_t)chunk * PIX + a * HW + t] = sum;
    }
}

__global__ void reduce_ev1(const float* __restrict__ part, float* __restrict__ ev1) {
    int i = blockIdx.x * 256 + threadIdx.x;
    if (i < PIX)
        ev1[i] = part[i] + part[PIX + i] + part[2 * PIX + i] + part[3 * PIX + i];
}

__global__ void pack_xd(const float* __restrict__ dual, const float* __restrict__ ev1,
                        const float* __restrict__ proj, float* __restrict__ xd) {
    int i = blockIdx.x * 256 + threadIdx.x;
    if (i < PIX) {
        xd[i]           = dual[i];
        xd[PIX + i]     = ev1[i];
        xd[2 * PIX + i] = proj[i];
    }
}

// ---------------------------------------------------------------------------
// Back projection: bp[p] = sum_{a,x} S[a,p,x] * dual[a,x]
// Block handles 12 consecutive p; entire dual (36 KB) staged in LDS.
// 2-way unroll over a with dual accumulators.
// ---------------------------------------------------------------------------
__global__ void gemv_back(const float* __restrict__ S,
                          const float* __restrict__ dual,
                          float* __restrict__ bp) {
    __shared__ float dl[PIX];
    __shared__ float red[288];
    const int t = threadIdx.x;
    for (int idx = t; idx < PIX; idx += 288) dl[idx] = dual[idx];
    __syncthreads();
    const int xg = t % 24, pl = t / 24;
    const int x0 = xg * 4;
    const int p = blockIdx.x * 12 + pl;
    float acc0 = 0.f, acc1 = 0.f;
    for (int a = 0; a < HW; a += 2) {
        v4f s0 = __builtin_nontemporal_load(
            (const v4f*)(S + ((size_t)a * PIX + p) * HW + x0));
        v4f s1 = __builtin_nontemporal_load(
            (const v4f*)(S + ((size_t)(a + 1) * PIX + p) * HW + x0));
        const float* d0 = &dl[a * HW + x0];
        const float* d1 = &dl[(a + 1) * HW + x0];
        acc0 += s0.x * d0[0] + s0.y * d0[1] + s0.z * d0[2] + s0.w * d0[3];
        acc1 += s1.x * d1[0] + s1.y * d1[1] + s1.z * d1[2] + s1.w * d1[3];
    }
    red[t] = acc0 + acc1;   // t = pl*24 + xg
    __syncthreads();
    if (t < 12) {
        float sum = 0.f;
        #pragma unroll
        for (int i = 0; i < 24; ++i) sum += red[t * 24 + i];
        bp[blockIdx.x * 12 + t] = sum;
    }
}

__global__ void pack_xp(const float* __restrict__ psq, const float* __restrict__ bp,
                        float* __restrict__ xp) {
    int i = blockIdx.x * 256 + threadIdx.x;
    if (i < 5 * PIX) xp[i] = psq[i];
    if (i < PIX) {
        int h = i / HW, w = i % HW;
        xp[5 * PIX + i] = bp[w * HW + h];   // ev2[h,w] = bp[w*96+h]
    }
}

// ---------------------------------------------------------------------------
// 3x3 SAME conv as implicit GEMM with V_WMMA_F32_16X16X4_F32 (full fp32).
// K ordered tap-major / channel-minor, channels padded to CP = ceil4(Cin).
// LDS layouts give each lane its v2f A/B fragment as one aligned ds_load_b64:
//   tin[(y'*98 + x')*CP + ci]   (3-row strip, halo + pad channels zeroed)
//   twt[(tap*32 + co)*CP + ci]  (weights transposed, pad co/ci zeroed)
// Steady-state inner loop: 2x ds_load_b64 + 1x v_wmma, no branches, full EXEC.
// Epilogue is compile-time specialized (ACT=PReLU, RES=residual add) and
// vectorized: each lane's 8 accumulators are 8 consecutive pixels -> 2x B128.
// ---------------------------------------------------------------------------
template <int CIN, int COUT, bool ACT, bool RES>
__global__ void conv3x3_wmma(const float* __restrict__ in,
                             const float* __restrict__ wgt,
                             const float* __restrict__ bias,
                             const float* __restrict__ alpha,
                             const float* __restrict__ residual,
                             float* __restrict__ out) {
    constexpr int CP = (CIN + 3) & ~3;
    __shared__ float tin[3 * 98 * CP];
    __shared__ float twt[9 * 32 * CP];
    const int tid = threadIdx.x;        // 0..63
    const int h   = blockIdx.x;         // output row

    // zero-fill (covers halo columns, OOB rows, channel padding, unused co)
    for (int i = tid; i < 3 * 98 * CP; i += 64) tin[i] = 0.f;
    for (int i = tid; i < 9 * 32 * CP; i += 64) twt[i] = 0.f;
    __syncthreads();

    // stage input strip rows h-1..h+1 (coalesced over x)
    for (int i = tid; i < CIN * 3 * HW; i += 64) {
        int ci = i / (3 * HW);
        int r  = i % (3 * HW);
        int yy = r / HW;                // 0..2
        int x  = r % HW;
        int y  = h - 1 + yy;
        if ((unsigned)y < (unsigned)HW)
            tin[(yy * 98 + x + 1) * CP + ci] = in[(size_t)ci * PIX + y * HW + x];
    }
    // stage weights (global reads are linear over wgt[co][ci][tap])
    for (int i = tid; i < COUT * CIN * 9; i += 64) {
        int co  = i / (CIN * 9);
        int r   = i % (CIN * 9);
        int ci  = r / 9;
        int tap = r % 9;
        twt[(tap * 32 + co) * CP + ci] = wgt[i];
    }
    __syncthreads();

    const int wave = tid >> 5;
    const int lane = tid & 31;
    const int half = lane >> 4;
    const int l16  = lane & 15;
    constexpr int NT = (COUT + 15) / 16;
    if (wave < NT) {
        const int co = wave * 16 + l16;
        const float al = ACT ? alpha[0] : 0.f;
        const float bs = (co < COUT) ? bias[co] : 0.f;
        for (int wt = 0; wt < 6; ++wt) {
            const int w0 = wt * 16;
            v8f c = {0.f, 0.f, 0.f, 0.f, 0.f, 0.f, 0.f, 0.f};
            #pragma unroll
            for (int tap = 0; tap < 9; ++tap) {
                const int dy = tap / 3, dx = tap % 3;
                // x = w0 + m + dx - 1  ->  x' = x + 1 = w0 + m + dx
                const float* ta = &tin[(dy * 98 + w0 + l16 + dx) * CP + 2 * half];
                const float* tb = &twt[(tap * 32 + co) * CP + 2 * half];
                #pragma unroll
                for (int c4 = 0; c4 < CP / 4; ++c4) {
                    v2f av = *(const v2f*)(ta + c4 * 4);
                    v2f bv = *(const v2f*)(tb + c4 * 4);
                    c = __builtin_amdgcn_wmma_f32_16x16x4_f32(
                        false, av, false, bv, (short)0, c, false, false);
                }
            }
            if (co < COUT) {
                // lane covers 8 consecutive pixels: base = w0 + half*8
                const size_t off = (size_t)co * PIX + h * HW + w0 + half * 8;
                v4f r0, r1;
                #pragma unroll
                for (int i = 0; i < 4; ++i) {
                    r0[i] = c[i] + bs;
                    r1[i] = c[4 + i] + bs;
                }
                if (ACT) {
                    #pragma unroll
                    for (int i = 0; i < 4; ++i) {
                        r0[i] = (r0[i] >= 0.f) ? r0[i] : al * r0[i];
                        r1[i] = (r1[i] >= 0.f) ? r1[i] : al * r1[i];
                    }
                }
                if (RES) {
                    const v4f* rp = (const v4f*)(residual + off);
                    r0 += rp[0];
                    r1 += rp[1];
                }
                v4f* op = (v4f*)(out + off);
                op[0] = r0;
                op[1] = r1;
            }
        }
    }
}

// ---------------------------------------------------------------------------
extern "C" void kernel_launch(void* const* d_in, const int* in_sizes, int n_in,
                              void* d_out, int out_size, void* d_ws, size_t ws_size,
                              hipStream_t stream) {
    (void)in_sizes; (void)n_in; (void)out_size; (void)ws_size;
    const float* in_dual   = (const float*)d_in[0];
    const float* in_primal = (const float*)d_in[1];
    const float* proj      = (const float*)d_in[2];
    const float* S         = (const float*)d_in[3];
    const float* S_norm    = (const float*)d_in[4];
    const float* dW1 = (const float*)d_in[5];
    const float* db1 = (const float*)d_in[6];
    const float* da1 = (const float*)d_in[7];
    const float* dW2 = (const float*)d_in[8];
    const float* db2 = (const float*)d_in[9];
    const float* da2 = (const float*)d_in[10];
    const float* dW3 = (const float*)d_in[11];
    const float* db3 = (const float*)d_in[12];
    const float* pW1 = (const float*)d_in[13];
    const float* pb1 = (const float*)d_in[14];
    const float* pa1 = (const float*)d_in[15];
    const float* pW2 = (const float*)d_in[16];
    const float* pb2 = (const float*)d_in[17];
    const float* pa2 = (const float*)d_in[18];
    const float* pW3 = (const float*)d_in[19];
    const float* pb3 = (const float*)d_in[20];

    float* ws   = (float*)d_ws;
    float* dual = ws;                 // 9216
    float* psq  = dual + PIX;         // 5*9216
    float* ev1  = psq + 5 * PIX;      // 9216
    float* ev1p = ev1 + PIX;          // 4*9216
    float* bp   = ev1p + 4 * PIX;     // 9216
    float* xd   = bp + PIX;           // 3*9216
    float* xp   = xd + 3 * PIX;       // 6*9216
    float* t1   = xp + 6 * PIX;       // 32*9216
    float* t2   = t1 + 32 * PIX;      // 32*9216

    hipMemcpyAsync(dual, in_dual, PIX * sizeof(float), hipMemcpyDeviceToDevice, stream);
    hipMemcpyAsync(psq, in_primal, 5 * PIX * sizeof(float), hipMemcpyDeviceToDevice, stream);

    for (int k = 0; k < 10; ++k) {
        // ---- dual update ----
        gemv_forward<<<dim3(96, 4), 288, 0, stream>>>(S_norm, psq, ev1p);
        reduce_ev1<<<36, 256, 0, stream>>>(ev1p, ev1);
        pack_xd<<<36, 256, 0, stream>>>(dual, ev1, proj, xd);
        conv3x3_wmma<3, 32, true, false><<<96, 64, 0, stream>>>(
            xd, dW1 + (size_t)k * 864, db1 + k * 32, da1 + k, nullptr, t1);
        conv3x3_wmma<32, 32, true, false><<<96, 64, 0, stream>>>(
            t1, dW2 + (size_t)k * 9216, db2 + k * 32, da2 + k, nullptr, t2);
        conv3x3_wmma<32, 1, false, true><<<96, 64, 0, stream>>>(
            t2, dW3 + (size_t)k * 288, db3 + k, nullptr, dual, dual);
        // ---- primal update ----
        gemv_back<<<768, 288, 0, stream>>>(S, dual, bp);
        pack_xp<<<180, 256, 0, stream>>>(psq, bp, xp);
        conv3x3_wmma<6, 32, true, false><<<96, 64, 0, stream>>>(
            xp, pW1 + (size_t)k * 1728, pb1 + k * 32, pa1 + k, nullptr, t1);
        conv3x3_wmma<32, 32, true, false><<<96, 64, 0, stream>>>(
            t1, pW2 + (size_t)k * 9216, pb2 + k * 32, pa2 + k, nullptr, t2);
        conv3x3_wmma<32, 5, false, true><<<96, 64, 0, stream>>>(
            t2, pW3 + (size_t)k * 1440, pb3 + k * 5, nullptr, psq, psq);
    }
    // output = primal[:, 0:1, :, :] == channel 0 of psq
    hipMemcpyAsync(d_out, psq, PIX * sizeof(float), hipMemcpyDeviceToDevice, stream);
}